// HpNsynth_40965398069324
// MI455X (gfx1250) — compile-verified
//
#include <hip/hip_runtime.h>
#include <math.h>

// ---------------- problem constants ----------------
#define SR_F     16000.0f
#define BLOCK    160
#define NFRAMES  1000
#define BATCH    4
#define NH       64
#define NB       65
#define REV_LEN  16000
#define TLEN     160000          // NFRAMES * BLOCK
#define OUTSEC   640000          // BATCH * TLEN
#define TWO_PI   6.28318530717958647692f

// ---------------- workspace layout (floats) ----------------
#define IMP_ALLOC   16384        // 16000 taps + zero slack (group-reversed)
#define WS_IMP      0
#define WS_S        16384        // 4000 per-frame phase sums
#define WS_O        20384        // 4000 per-frame phase offsets
#define WS_XPAD     24384        // 4 * XPAD_STRIDE zero-padded dereverb
#define XPAD_PAD    16384        // left zero pad (16000 taps + prefetch slack)
#define XPAD_STRIDE 176384       // XPAD_PAD + TLEN
// total = 24384 + 4*176384 = 729,920 floats = 2.92 MB

typedef __attribute__((ext_vector_type(2))) float v2f;
typedef __attribute__((ext_vector_type(8))) float v8f;

#if __has_builtin(__builtin_amdgcn_sched_group_barrier)
#define SGB(mask, size, id) __builtin_amdgcn_sched_group_barrier(mask, size, id)
#else
#define SGB(mask, size, id)
#endif

__device__ __forceinline__ float expsig(float x) {
  // 2 * sigmoid(x)^ln(10) + 1e-7
  float s = 1.0f / (1.0f + __expf(-x));
  return 2.0f * __powf(s, 2.302585092994046f) + 1e-7f;
}

__device__ __forceinline__ float f0_upsample(const float* f0b, int t) {
  // torch linear interp, align_corners=False, factor 160
  float pos = (t + 0.5f) * (1.0f / (float)BLOCK) - 0.5f;
  pos = fminf(fmaxf(pos, 0.0f), (float)(NFRAMES - 1));
  int i0 = (int)pos;
  int i1 = min(i0 + 1, NFRAMES - 1);
  float frac = pos - (float)i0;
  return f0b[i0] * (1.0f - frac) + f0b[i1] * frac;
}

// ---------------- K0: reverb impulse (group-of-4 reversed) + zero pads ------
__global__ void k0_impulse(const float* __restrict__ rnoise,
                           const float* __restrict__ decay,
                           const float* __restrict__ wet,
                           float* __restrict__ ws) {
  int j = blockIdx.x * blockDim.x + threadIdx.x;
  if (j >= IMP_ALLOC) return;
  float v = 0.0f;
  if (j < REV_LEN) {
    float z = -decay[0];
    float sp = fmaxf(z, 0.0f) + log1pf(__expf(-fabsf(z)));   // softplus(-decay)
    float wsig = 1.0f / (1.0f + __expf(-wet[0]));
    float tj = (float)j / SR_F;
    v = rnoise[j] * __expf(-sp * tj * 500.0f) * wsig;
    if (j == 0) v = 1.0f;
  }
  // store group-of-4 reversed: G[4g + r] = imp[4g + 3 - r]
  ws[WS_IMP + (j & ~3) + (3 - (j & 3))] = v;
  for (int b = 0; b < BATCH; ++b)
    ws[WS_XPAD + b * XPAD_STRIDE + j] = 0.0f;
}

// ---------------- K1: per-frame phase-increment sums ----------------
__global__ void k1_framesum(const float* __restrict__ f0, float* __restrict__ ws) {
  __shared__ float sw[BLOCK];
  int f = blockIdx.x, b = blockIdx.y, tid = threadIdx.x;
  const float* f0b = f0 + b * NFRAMES;
  sw[tid] = TWO_PI * f0_upsample(f0b, f * BLOCK + tid) / SR_F;
  __syncthreads();
  if (tid == 0) {
    float s = 0.0f;
    for (int i = 0; i < BLOCK; ++i) s += sw[i];
    ws[WS_S + b * NFRAMES + f] = s;
  }
}

// ---------------- K2: exclusive prefix over frames (tiny) ----------------
__global__ void k2_prefix(float* __restrict__ ws) {
  int b = threadIdx.x;
  if (b >= BATCH) return;
  float run = 0.0f;
  for (int f = 0; f < NFRAMES; ++f) {
    ws[WS_O + b * NFRAMES + f] = run;
    run += ws[WS_S + b * NFRAMES + f];
  }
}

// ---------------- K3: fused per-frame synth (harmonic + filtered noise) ----
__global__ void __launch_bounds__(BLOCK)
k3_synth(const float* __restrict__ amp, const float* __restrict__ hd,
         const float* __restrict__ nb, const float* __restrict__ f0,
         const float* __restrict__ nunit, float* __restrict__ out,
         float* __restrict__ ws) {
  __shared__ float sh_h[NH];
  __shared__ float sh_a[NB];
  __shared__ float sh_ir[128];
  __shared__ float sh_ker[BLOCK];
  __shared__ float sh_ns[BLOCK];
  __shared__ float sh_scan[BLOCK];
  __shared__ float sh_scale;

  int f = blockIdx.x, b = blockIdx.y, tid = threadIdx.x;
  int fr = b * NFRAMES + f;
  float f0v = f0[fr];

  if (tid < NB) sh_a[tid] = expsig(nb[fr * NB + tid]);
  if (tid < NH) {
    float hk = expsig(hd[fr * NH + tid]);
    float aa = ((f0v * (float)(tid + 1) < 8000.0f) ? 1.0f : 0.0f) + 1e-4f;
    sh_h[tid] = hk * aa;
  }
  __syncthreads();
  if (tid == 0) {
    float s = 0.0f;
    for (int k = 0; k < NH; ++k) s += sh_h[k];
    sh_scale = expsig(amp[fr]) / s;
  }
  __syncthreads();
  if (tid < NH) sh_h[tid] *= sh_scale;

  // inclusive scan of phase increments within the frame
  const float* f0b = f0 + b * NFRAMES;
  sh_scan[tid] = TWO_PI * f0_upsample(f0b, f * BLOCK + tid) / SR_F;
  __syncthreads();
  for (int off = 1; off < BLOCK; off <<= 1) {
    float v = (tid >= off) ? sh_scan[tid - off] : 0.0f;
    __syncthreads();
    sh_scan[tid] += v;
    __syncthreads();
  }
  float omega = ws[WS_O + fr] + sh_scan[tid];
  float om = fmodf(omega, TWO_PI);   // sin(k*om) == sin(k*omega) for integer k

  float hs = 0.0f;
  for (int k = 0; k < NH; ++k)
    hs += __sinf(om * (float)(k + 1)) * sh_h[k];

  // 65-bin zero-phase irfft -> 128-tap IR
  if (tid < 128) {
    float n = (float)tid;
    float acc = sh_a[0] + sh_a[NB - 1] * ((tid & 1) ? -1.0f : 1.0f);
    for (int q = 1; q < NB - 1; ++q)
      acc += 2.0f * sh_a[q] * __cosf(TWO_PI * (float)q * n * (1.0f / 128.0f));
    sh_ir[tid] = acc * (1.0f / 128.0f);
  }
  sh_ns[tid] = nunit[fr * BLOCK + tid] * 2.0f - 1.0f;
  __syncthreads();

  // roll(+64), Hann, pad to 160, roll(-64)  ==>  rearranged windowed kernel
  float kv = 0.0f;
  if (tid < 64)
    kv = sh_ir[tid] * (0.5f - 0.5f * __cosf(TWO_PI * (float)(tid + 64) * (1.0f / 128.0f)));
  else if (tid >= 96)
    kv = sh_ir[tid - 32] * (0.5f - 0.5f * __cosf(TWO_PI * (float)(tid - 96) * (1.0f / 128.0f)));
  sh_ker[tid] = kv;
  __syncthreads();

  // fft_convolve(noise, ker) == causal linear conv truncated to frame
  float nv = 0.0f;
  for (int m = 0; m <= tid; ++m) nv += sh_ns[m] * sh_ker[tid - m];

  int gi = b * TLEN + f * BLOCK + tid;
  float dv = hs + nv;
  out[1 * OUTSEC + gi] = dv;   // dereverb
  out[2 * OUTSEC + gi] = nv;   // noise
  out[3 * OUTSEC + gi] = hs;   // harmonic_signal
  ws[WS_XPAD + b * XPAD_STRIDE + XPAD_PAD + f * BLOCK + tid] = dv;
}

// ---------------- K4: 16000-tap reverb via V_WMMA_F32_16X16X4_F32 ----------
// Tap within a K-group is reversed: j = j0 + 4n + (3-k), so
//   A[m,k] = x[i0 + m - 3 + k - j0]      (ascending in k -> natural b64 load)
//   B[k,n] = imp[j0 + 4n + 3 - k] = G[j0 + 4n + k]  (G = group-4-reversed imp)
// C[m,n] = partial of y[i0+m+4n] over tap class n; 100% MAC utilization.
// Inner loop: 4 rotating buffer pairs, unroll x4, prefetch one iteration
// ahead; sched_group_barrier pins [wmma x2][load x4][wmma x2][load x4] so
// 4 loads stay in flight across every WMMA pair (s_wait_loadcnt > 0).
#define TILE_CHUNK 25
#define NTILES     10000         // TLEN / 16
#define K4_ITERS   63            // 252 WMMAs; taps 16000..16127 are zeros

__global__ void __launch_bounds__(128)
k4_reverb_wmma(const float* __restrict__ ws_imp,
               const float* __restrict__ ws_xpad,
               float* __restrict__ synth) {
  __shared__ float ldsC[4][5 * 256];     // 4 waves x 5-slot C ring
  int wave = threadIdx.x >> 5;
  int lane = threadIdx.x & 31;
  int b = blockIdx.y;
  int chunk = blockIdx.x * 4 + wave;
  int tstart = chunk * TILE_CHUNK;
  const float* xb = ws_xpad + b * XPAD_STRIDE;   // index 0 == x[-XPAD_PAD]
  float* myC = &ldsC[wave][0];

  // A: lanes 0-15 -> M=lane, K={0,1}; lanes 16-31 -> M=lane-16, K={2,3}
  int ea = (lane < 16) ? (lane - 3) : (lane - 17);
  // B: column n = lane&15; lanes 0-15 -> K={0,1}; lanes 16-31 -> K={2,3}
  int nIdx = lane & 15;
  const float* bbase = ws_imp + 4 * nIdx + ((lane >> 4) << 1);

  for (int t = tstart - 4; t < tstart + TILE_CHUNK; ++t) {
    int slot = ((t % 5) + 5) % 5;
    if (t < 0) {
      for (int r = 0; r < 8; ++r) myC[slot * 256 + r * 32 + lane] = 0.0f;
    } else {
      v8f c = {0.0f, 0.0f, 0.0f, 0.0f, 0.0f, 0.0f, 0.0f, 0.0f};
      const float* pa0 = xb + (XPAD_PAD + t * 16 + ea);   // j0 = 0
      const float* pa1 = pa0 - 64;                        // j0 = 64
      const float* pa2 = pa0 - 128;                       // j0 = 128
      const float* pa3 = pa0 - 192;                       // j0 = 192
      const float* pb0 = bbase;
      const float* pb1 = bbase + 64;
      const float* pb2 = bbase + 128;
      const float* pb3 = bbase + 192;
      v2f a0 = *(const v2f*)pa0;  v2f b0 = *(const v2f*)pb0;
      v2f a1 = *(const v2f*)pa1;  v2f b1 = *(const v2f*)pb1;
      v2f a2 = *(const v2f*)pa2;  v2f b2 = *(const v2f*)pb2;
      v2f a3 = *(const v2f*)pa3;  v2f b3 = *(const v2f*)pb3;
      for (int it = 0; it < K4_ITERS; ++it) {
        c = __builtin_amdgcn_wmma_f32_16x16x4_f32(false, a0, false, b0,
                                                  (short)0, c, false, false);
        c = __builtin_amdgcn_wmma_f32_16x16x4_f32(false, a1, false, b1,
                                                  (short)0, c, false, false);
        pa0 -= 256; pb0 += 256;
        a0 = *(const v2f*)pa0;  b0 = *(const v2f*)pb0;
        pa1 -= 256; pb1 += 256;
        a1 = *(const v2f*)pa1;  b1 = *(const v2f*)pb1;
        c = __builtin_amdgcn_wmma_f32_16x16x4_f32(false, a2, false, b2,
                                                  (short)0, c, false, false);
        c = __builtin_amdgcn_wmma_f32_16x16x4_f32(false, a3, false, b3,
                                                  (short)0, c, false, false);
        pa2 -= 256; pb2 += 256;
        a2 = *(const v2f*)pa2;  b2 = *(const v2f*)pb2;
        pa3 -= 256; pb3 += 256;
        a3 = *(const v2f*)pa3;  b3 = *(const v2f*)pb3;
        // pin schedule: 2 WMMA, 4 vmem-reads, 2 WMMA, 4 vmem-reads
        SGB(0x008, 2, 0);
        SGB(0x020, 4, 0);
        SGB(0x008, 2, 0);
        SGB(0x020, 4, 0);
      }
      int mbase = (lane >> 4) * 8;       // C/D: VGPR r holds M = r + 8*(lane>=16)
      for (int r = 0; r < 8; ++r)
        myC[slot * 256 + (mbase + r) * 16 + nIdx] = c[r];
    }
    __syncthreads();
    if (t >= tstart && lane < 16) {
      int i = t * 16 + lane;
      float y = 0.0f;
      for (int n = 0; n < 16; ++n) {
        int idx = i - 4 * n;
        int m = idx & 15;
        int tau = idx >> 4;              // arithmetic shift: floor, may be -1..-4
        int sl = ((tau % 5) + 5) % 5;    // negative tau slots were zero-filled
        y += myC[sl * 256 + m * 16 + n];
      }
      synth[b * TLEN + i] = y;
    }
    __syncthreads();
  }
}

// ---------------- launch ----------------
extern "C" void kernel_launch(void* const* d_in, const int* in_sizes, int n_in,
                              void* d_out, int out_size, void* d_ws, size_t ws_size,
                              hipStream_t stream) {
  (void)in_sizes; (void)n_in; (void)out_size; (void)ws_size;
  const float* amp = (const float*)d_in[0];   // [B,F,1]
  const float* hd  = (const float*)d_in[1];   // [B,F,64]
  const float* nb  = (const float*)d_in[2];   // [B,F,65]
  const float* f0  = (const float*)d_in[3];   // [B,F,1]
  const float* nu  = (const float*)d_in[4];   // [B,F,160]
  const float* rn  = (const float*)d_in[5];   // [16000,1]
  const float* dec = (const float*)d_in[6];   // scalar
  const float* wet = (const float*)d_in[7];   // scalar
  float* out = (float*)d_out;                 // synth|dereverb|noise|harmonic, 4x640000
  float* ws  = (float*)d_ws;                  // needs >= 2,919,680 bytes

  k0_impulse<<<dim3(IMP_ALLOC / 256), dim3(256), 0, stream>>>(rn, dec, wet, ws);
  k1_framesum<<<dim3(NFRAMES, BATCH), dim3(BLOCK), 0, stream>>>(f0, ws);
  k2_prefix<<<dim3(1), dim3(32), 0, stream>>>(ws);
  k3_synth<<<dim3(NFRAMES, BATCH), dim3(BLOCK), 0, stream>>>(amp, hd, nb, f0, nu, out, ws);
  k4_reverb_wmma<<<dim3(NTILES / TILE_CHUNK / 4, BATCH), dim3(128), 0, stream>>>(
      ws + WS_IMP, ws + WS_XPAD, out);
}